// HierarchicalLoss_5437428597039
// MI455X (gfx1250) — compile-verified
//
#include <hip/hip_runtime.h>
#include <hip/hip_bf16.h>
#include <cstdint>
#include <cstddef>

// ---------------------------------------------------------------------------
// HierarchicalLoss on MI455X (gfx1250, wave32).
//
// Pass 1 (HBM-bound, ~178MB @ 23.3TB/s ~ 7.6us): wave-per-row coalesced
//   float4 stream -> row max (shfl_xor wave32 reduce) + x[row, target].
// Pass 2 (mostly L2-resident, 192MB L2 covers the 178MB of logits): wave per
//   16-row tile; exp(x - rowmax) fed as a 16x4 fp32 A-matrix into
//   V_WMMA_F32_16X16X4_F32 against an all-ones B, accumulating per-row sums
//   in the fp32 C/D registers (full precision, 64 elems reduced per issue).
// Pass 3: single block, fixed-order deterministic reduction of NLLs + the
//   integer dependency-miss count; writes (total, total_ce, total_dep).
// ---------------------------------------------------------------------------

typedef __attribute__((ext_vector_type(2))) float v2f;
typedef __attribute__((ext_vector_type(8))) float v8f;

#define NROWS      4096
#define TOTROWS    (4 * NROWS)          // 16384 rows across the 4 levels
#define EM1        1.7182818284590452f  // e - 1

// ---------------------------------------------------------------------------
// Kernel 1: per-row max and x[row, target].  One wave (32 lanes) per row.
// Coalesced: lane L reads float4 at [chunk*128 + 4L]. C in {128,512,2048,8192}
// is always a multiple of 128, so every wave-chunk is exactly 512B contiguous.
// ---------------------------------------------------------------------------
__global__ __launch_bounds__(256) void hl_rowmax_kernel(
    const float* __restrict__ o0, const float* __restrict__ o1,
    const float* __restrict__ o2, const float* __restrict__ o3,
    const int*  __restrict__ targets,   // [4096, 4] int32
    float* __restrict__ rowmax,         // [16384]
    float* __restrict__ xt)             // [16384]
{
    const int wave  = blockIdx.x * 8 + (threadIdx.x >> 5);  // global row id
    const int lane  = threadIdx.x & 31;
    const int level = wave >> 12;       // 0..3
    const int r     = wave & (NROWS - 1);

    const float* x; int C;
    if      (level == 0) { x = o0; C = 128;  }
    else if (level == 1) { x = o1; C = 512;  }
    else if (level == 2) { x = o2; C = 2048; }
    else                 { x = o3; C = 8192; }

    const float* rowp = x + (size_t)r * (size_t)C;
    const int iters = C >> 7;           // chunks of 128 floats (1..64)

    float m = -3.402823466e38f;
    for (int it = 0; it < iters; ++it) {
        const float* p = rowp + it * 128 + lane * 4;
        // speculative prefetch one chunk ahead -> global_prefetch_b8
        __builtin_prefetch(p + 128, 0, 0);
        const float4 v = *reinterpret_cast<const float4*>(p);
        m = fmaxf(m, fmaxf(fmaxf(v.x, v.y), fmaxf(v.z, v.w)));
    }
    // wave32 butterfly max
    for (int off = 16; off; off >>= 1)
        m = fmaxf(m, __shfl_xor(m, off, 32));

    if (lane == 0) {
        rowmax[wave] = m;
        const int t = targets[r * 4 + level];   // target class for this level
        xt[wave] = rowp[t];
    }
}

// ---------------------------------------------------------------------------
// Kernel 2: per-row sum of exp(x - rowmax) via V_WMMA_F32_16X16X4_F32.
// One wave per 16-row tile.  A-matrix (16x4 f32) layout per ISA:
//   lanes 0-15 : M=lane, VGPR0=K0, VGPR1=K1   -> cols c+0, c+1
//   lanes 16-31: M=lane-16, VGPR0=K2, VGPR1=K3 -> cols c+2, c+3
// B = ones => D[m,n] = sum_k A[m,k]; accumulating over all column blocks
// yields full row sums in C.  Four accumulators break the WMMA RAW chain;
// identical C/D layouts permit an element-wise combine at the end.
// Row sums: lane 0 VGPR j -> row r0+j (j=0..7); lane 16 VGPR j -> row r0+8+j.
// Each row consumes its 16-col chunk = exactly one 64B line: zero waste.
// ---------------------------------------------------------------------------
__global__ __launch_bounds__(128) void hl_sumexp_wmma_kernel(
    const float* __restrict__ o0, const float* __restrict__ o1,
    const float* __restrict__ o2, const float* __restrict__ o3,
    const float* __restrict__ rowmax,   // [16384]
    const float* __restrict__ xt,       // [16384]
    float* __restrict__ nll)            // [16384]
{
    const int g     = blockIdx.x * 4 + (threadIdx.x >> 5);  // tile id 0..1023
    const int lane  = threadIdx.x & 31;
    const int level = g >> 8;                               // 256 tiles/level
    const int r0    = (g & 255) << 4;                       // first row of tile
    const int mrow  = lane & 15;                            // row within tile
    const int half  = lane >> 4;                            // K-pair selector

    const float* x; int C;
    if      (level == 0) { x = o0; C = 128;  }
    else if (level == 1) { x = o1; C = 512;  }
    else if (level == 2) { x = o2; C = 2048; }
    else                 { x = o3; C = 8192; }

    const int    grow = (level << 12) + r0;                 // global row base
    const float  m    = rowmax[grow + mrow];                // this lane's row max
    const float* rp   = x + (size_t)(r0 + mrow) * (size_t)C + half * 2;

    v2f ones; ones[0] = 1.0f; ones[1] = 1.0f;
    v8f acc0 = {}; v8f acc1 = {}; v8f acc2 = {}; v8f acc3 = {};

    for (int c = 0; c < C; c += 16) {
        v2f a0 = *reinterpret_cast<const v2f*>(rp + c + 0);
        v2f a1 = *reinterpret_cast<const v2f*>(rp + c + 4);
        v2f a2 = *reinterpret_cast<const v2f*>(rp + c + 8);
        v2f a3 = *reinterpret_cast<const v2f*>(rp + c + 12);
        v2f e0, e1, e2, e3;
        e0[0] = __expf(a0[0] - m); e0[1] = __expf(a0[1] - m);
        e1[0] = __expf(a1[0] - m); e1[1] = __expf(a1[1] - m);
        e2[0] = __expf(a2[0] - m); e2[1] = __expf(a2[1] - m);
        e3[0] = __expf(a3[0] - m); e3[1] = __expf(a3[1] - m);
        // (neg_a, A, neg_b, B, c_mod, C, reuse_a, reuse_b)
        acc0 = __builtin_amdgcn_wmma_f32_16x16x4_f32(false, e0, false, ones,
                                                     (short)0, acc0, false, false);
        acc1 = __builtin_amdgcn_wmma_f32_16x16x4_f32(false, e1, false, ones,
                                                     (short)0, acc1, false, false);
        acc2 = __builtin_amdgcn_wmma_f32_16x16x4_f32(false, e2, false, ones,
                                                     (short)0, acc2, false, false);
        acc3 = __builtin_amdgcn_wmma_f32_16x16x4_f32(false, e3, false, ones,
                                                     (short)0, acc3, false, false);
    }

    // combine accumulators element-wise (same D layout)
    v8f acc = (acc0 + acc1) + (acc2 + acc3);

    if (lane == 0 || lane == 16) {
        const int base = grow + half * 8;
        #pragma unroll
        for (int j = 0; j < 8; ++j) {
            const int row = base + j;
            const float S = acc[j];                         // row sum of exp
            nll[row] = rowmax[row] + __logf(S) - xt[row];   // -(x_t - m - logS)
        }
    }
}

// ---------------------------------------------------------------------------
// Kernel 3: deterministic final reduction (single block, fixed pairing).
// total_ce  = sum(all 16384 nll) / 4096   (all levels share the denominator)
// total_dep = (e-1) * (# invalid parent-child pairs) / 4096
// ---------------------------------------------------------------------------
__global__ __launch_bounds__(256) void hl_finalize_kernel(
    const float* __restrict__ nll,
    const int* __restrict__ p0, const int* __restrict__ p1,
    const int* __restrict__ p2, const int* __restrict__ p3,
    const unsigned char* __restrict__ v01,   // [128, 512]  bool
    const unsigned char* __restrict__ v12,   // [512, 2048] bool
    const unsigned char* __restrict__ v23,   // [2048,8192] bool
    float* __restrict__ out)                 // [3]: total, ce, dep
{
    __shared__ float sf[256];
    __shared__ int   si[256];
    const int t = threadIdx.x;

    float s = 0.0f;
    for (int i = t; i < TOTROWS; i += 256) s += nll[i];

    int bad = 0;
    for (int i = t; i < NROWS; i += 256) {
        const int a = p0[i], b = p1[i], c = p2[i], d = p3[i];
        bad += (v01[(size_t)a * 512  + b] == 0);
        bad += (v12[(size_t)b * 2048 + c] == 0);
        bad += (v23[(size_t)c * 8192 + d] == 0);
    }

    sf[t] = s; si[t] = bad;
    __syncthreads();
    for (int off = 128; off; off >>= 1) {
        if (t < off) { sf[t] += sf[t + off]; si[t] += si[t + off]; }
        __syncthreads();
    }
    if (t == 0) {
        const float ce  = sf[0] * (1.0f / 4096.0f);
        const float dep = EM1 * (float)si[0] * (1.0f / 4096.0f);
        out[1] = ce;
        out[2] = dep;
        out[0] = 0.5f * ce + 0.5f * dep;
    }
}

// ---------------------------------------------------------------------------
// Launch
// ---------------------------------------------------------------------------
extern "C" void kernel_launch(void* const* d_in, const int* in_sizes, int n_in,
                              void* d_out, int out_size, void* d_ws, size_t ws_size,
                              hipStream_t stream) {
    const float* o0 = (const float*)d_in[0];
    const float* o1 = (const float*)d_in[1];
    const float* o2 = (const float*)d_in[2];
    const float* o3 = (const float*)d_in[3];
    const int*   tg = (const int*)  d_in[4];
    const int*   p0 = (const int*)  d_in[5];
    const int*   p1 = (const int*)  d_in[6];
    const int*   p2 = (const int*)  d_in[7];
    const int*   p3 = (const int*)  d_in[8];
    const unsigned char* v01 = (const unsigned char*)d_in[9];
    const unsigned char* v12 = (const unsigned char*)d_in[10];
    const unsigned char* v23 = (const unsigned char*)d_in[11];
    float* out = (float*)d_out;

    float* rowmax = (float*)d_ws;            // 16384 floats
    float* xt     = rowmax + TOTROWS;        // 16384 floats
    float* nll    = xt + TOTROWS;            // 16384 floats  (192KB total)

    // 16384 rows, 8 waves/block -> 2048 blocks
    hl_rowmax_kernel<<<TOTROWS / 8, 256, 0, stream>>>(o0, o1, o2, o3, tg,
                                                      rowmax, xt);
    // 1024 16-row tiles, 4 waves/block -> 256 blocks
    hl_sumexp_wmma_kernel<<<256, 128, 0, stream>>>(o0, o1, o2, o3,
                                                   rowmax, xt, nll);
    hl_finalize_kernel<<<1, 256, 0, stream>>>(nll, p0, p1, p2, p3,
                                              v01, v12, v23, out);
}